// MultiBoxLoss_47253230190629
// MI455X (gfx1250) — compile-verified
//
#include <hip/hip_runtime.h>
#include <stdint.h>

// ---------------- problem constants (from reference setup_inputs) -------------
constexpr int NCLS   = 81;
constexpr int BA     = 64 * 8732;        // 558848 anchors
constexpr int TILE_A = 16;               // anchors per wave tile
constexpr int TILES  = BA / TILE_A;      // 34928 (exact)
constexpr int WPB    = 8;                // waves per block (256 threads, wave32)
constexpr int K1_BLOCKS = TILES / WPB;   // 4366 (exact)
constexpr int K3_BLOCKS = BA / 256;      // 2183 (exact)
constexpr int TILE_F = TILE_A * NCLS;    // 1296 floats = 5184 bytes
constexpr int TILE_CHUNKS = TILE_F / 4;  // 324 x b128 chunks (exact)
constexpr int HBINS = 2048;

typedef float v2f __attribute__((ext_vector_type(2)));
typedef float v8f __attribute__((ext_vector_type(8)));

// Generic pointer to a __shared__ object: low 32 bits are the wave-relative
// LDS byte offset (aperture form), which is what async-to-LDS's VDST wants.
__device__ __forceinline__ unsigned lds_off_u32(const void* p) {
  return (unsigned)(uintptr_t)p;
}

__device__ __forceinline__ void async_b128(unsigned lds_off, const float* g) {
  asm volatile("global_load_async_to_lds_b128 %0, %1, off"
               :: "v"(lds_off), "v"(g) : "memory");
}

__device__ __forceinline__ float sl1(float d) {
  d = fabsf(d);
  return d < 1.0f ? 0.5f * d * d : d - 0.5f;
}

// Fixed-topology wave32 reduction (deterministic).
__device__ __forceinline__ float wave_sum(float v) {
#pragma unroll
  for (int d = 16; d > 0; d >>= 1) v += __shfl_down(v, d, 32);
  return v;  // valid in lane 0
}

// ---------------- K0: zero the accumulated scratch regions -------------------
__global__ void k0_init(unsigned* g_hist, unsigned* g_sub, unsigned* scal) {
  int i = blockIdx.x * 256 + threadIdx.x;
  if (i < HBINS) { g_hist[i] = 0u; g_sub[i] = 0u; }
  if (i < 16) scal[i] = 0u;
}

// ---------------- K1: CE per anchor (WMMA softmax-sum) + loc loss + histogram -
__global__ __launch_bounds__(256) void k1_main(
    const float* __restrict__ pred_loc, const float* __restrict__ pred_conf,
    const float* __restrict__ gt_loc, const int* __restrict__ gt_label,
    float* __restrict__ ce_buf, unsigned* __restrict__ g_hist,
    float* __restrict__ partA) {
  __shared__ float    s_tile[WPB * TILE_F + 8];  // +pad: masked overshoot reads
  __shared__ float    s_sums[WPB * TILE_A];
  __shared__ unsigned s_hist[HBINS];
  __shared__ float    s_pl[WPB], s_pc[WPB], s_pn[WPB];

  const int tid  = threadIdx.x;
  const int lane = tid & 31;
  const int wave = tid >> 5;
  const int tile = blockIdx.x * WPB + wave;  // < TILES by construction

  // ---- issue async stream first: 5184B tile (16 anchors x 81 cls) -> LDS ---
  const float* gsrc = pred_conf + (size_t)tile * TILE_F;  // 16B-aligned
  const unsigned lds_base = lds_off_u32(&s_tile[wave * TILE_F]);
#pragma unroll
  for (int it = 0; it < 10; ++it) {                 // 324 = 10*32 + 4 chunks
    const int k = lane + it * 32;
    async_b128(lds_base + (unsigned)(k * 16), gsrc + k * 4);
  }
  if (lane < 4) {
    const int k = 320 + lane;
    async_b128(lds_base + (unsigned)(k * 16), gsrc + k * 4);
  }

  // overlap the copy with histogram zeroing
  for (int i = tid; i < HBINS; i += 256) s_hist[i] = 0u;
  __syncthreads();
  asm volatile("s_wait_asynccnt 0" ::: "memory");

  // ---- sum_c exp(x) via chained V_WMMA_F32_16X16X4_F32, B = ones -----------
  // A layout: lanes 0-15 hold row m, K={0,1}; lanes 16-31 hold row m, K={2,3}.
  const int row    = lane & 15;
  const int colsel = (lane >> 4) << 1;  // 0 or 2
  const float* tp  = &s_tile[wave * TILE_F + row * NCLS];
  v8f acc = {0.f, 0.f, 0.f, 0.f, 0.f, 0.f, 0.f, 0.f};
  v2f bones; bones.x = 1.0f; bones.y = 1.0f;
#pragma unroll
  for (int j = 0; j < 21; ++j) {        // 21*4 = 84 >= 81 classes
    const int col = 4 * j + colsel;
    float e0 = __expf(tp[col]);
    float e1 = __expf(tp[col + 1]);
    if (j == 20) {                      // classes 81..83 are padding -> 0
      e1 = 0.f;
      e0 = colsel ? 0.f : e0;
    }
    v2f a; a.x = e0; a.y = e1;
    acc = __builtin_amdgcn_wmma_f32_16x16x4_f32(false, a, false, bones,
                                                (short)0, acc, false, false);
  }
  // D column N=0 lives in lane 0 (rows 0..7) and lane 16 (rows 8..15).
  if (lane == 0 || lane == 16) {
    const int s = wave * TILE_A + (lane ? 8 : 0);
#pragma unroll
    for (int r = 0; r < 8; ++r) s_sums[s + r] = acc[r];
  }
  asm volatile("s_wait_dscnt 0" ::: "memory");  // same-wave LDS RAW fence

  // ---- per-anchor tail: ce, loc loss, histogram ----------------------------
  float locL = 0.f, posCe = 0.f, posN = 0.f;
  if (lane < TILE_A) {
    const int   anchor = tile * TILE_A + lane;
    const float sumexp = s_sums[wave * TILE_A + lane];
    const int   label  = gt_label[anchor];
    const float x      = s_tile[wave * TILE_F + lane * NCLS + label];
    const float ce     = __logf(sumexp) - x;  // = logsumexp - x[label], ce >= 0
    const bool  pos    = label > 0;
    ce_buf[anchor] = pos ? -1.0f : ce;        // sentinel excludes positives
    if (pos) {
      posN = 1.f; posCe = ce;
      const float4 pl = ((const float4*)pred_loc)[anchor];
      const float4 gl = ((const float4*)gt_loc)[anchor];
      locL = sl1(pl.x - gl.x) + sl1(pl.y - gl.y) +
             sl1(pl.z - gl.z) + sl1(pl.w - gl.w);
    } else {
      atomicAdd(&s_hist[__float_as_uint(ce) >> 21], 1u);  // monotone bins
    }
  }
  locL = wave_sum(locL); posCe = wave_sum(posCe); posN = wave_sum(posN);
  if (lane == 0) { s_pl[wave] = locL; s_pc[wave] = posCe; s_pn[wave] = posN; }
  __syncthreads();
  if (tid == 0) {
    float a = 0.f, b = 0.f, c = 0.f;
#pragma unroll
    for (int w = 0; w < WPB; ++w) { a += s_pl[w]; b += s_pc[w]; c += s_pn[w]; }
    partA[blockIdx.x * 3 + 0] = a;
    partA[blockIdx.x * 3 + 1] = b;
    partA[blockIdx.x * 3 + 2] = c;
  }
  for (int i = tid; i < HBINS; i += 256) {
    unsigned v = s_hist[i];
    if (v) atomicAdd(&g_hist[i], v);  // u32: order-independent, deterministic
  }
}

// ---------------- K2: reduce partials, pick coarse threshold bin T -----------
// Parallel but deterministic: fixed strided assignment, fixed fold order,
// integer suffix-scan for the threshold search.
__global__ __launch_bounds__(256) void k2_select(
    const float* __restrict__ partA, const unsigned* __restrict__ g_hist,
    unsigned* __restrict__ scal_u, float* __restrict__ scal_f) {
  __shared__ float    r0[256], r1[256], r2[256];
  __shared__ unsigned s_cnt[HBINS];
  __shared__ int      s_suf[256];
  __shared__ float    s_np;
  __shared__ unsigned s_T, s_R;
  const int t = threadIdx.x;

  float ll = 0.f, pc = 0.f, np = 0.f;
  for (int b = t; b < K1_BLOCKS; b += 256) {
    ll += partA[3 * b + 0]; pc += partA[3 * b + 1]; np += partA[3 * b + 2];
  }
  r0[t] = ll; r1[t] = pc; r2[t] = np;
  for (int i = t; i < HBINS; i += 256) s_cnt[i] = g_hist[i];
  __syncthreads();

  int csum = 0;  // thread t owns bins [8t, 8t+8)
#pragma unroll
  for (int i = 0; i < 8; ++i) csum += (int)s_cnt[8 * t + i];
  s_suf[t] = csum;
  __syncthreads();

  if (t == 0) {
    float a = 0.f, b = 0.f, c = 0.f;
    for (int i = 0; i < 256; ++i) { a += r0[i]; b += r1[i]; c += r2[i]; }
    scal_f[0] = c; scal_f[1] = a; scal_f[2] = b;  // np, loc, pos_ce
    s_np = c;
    int suf = 0;  // s_suf[i] := sum of chunk totals strictly above chunk i
    for (int i = 255; i >= 0; --i) { int cs = s_suf[i]; s_suf[i] = suf; suf += cs; }
    s_T = 0xFFFFu; s_R = 0u;
  }
  __syncthreads();

  const int npi = (int)s_np;
  int K = 3 * npi;
  const int mx = BA - npi;
  if (K > mx) K = mx;
  if (K > 0) {
    const int above = s_suf[t];
    if (above < K && above + csum >= K) {  // crossing lies in this chunk
      int cum = above;
      for (int i = 7; i >= 0; --i) {
        const int c = (int)s_cnt[8 * t + i];
        if (cum + c >= K) { s_T = (unsigned)(8 * t + i); s_R = (unsigned)(K - cum); break; }
        cum += c;
      }
    }
  }
  __syncthreads();
  if (t == 0) { scal_u[0] = s_T; scal_u[1] = s_R; }
}

// ---------------- K3: sum CE strictly above bin T; sub-histogram of bin T ----
__global__ __launch_bounds__(256) void k3_above(
    const float* __restrict__ ce_buf, const unsigned* __restrict__ scal_u,
    unsigned* __restrict__ g_sub, float* __restrict__ partB) {
  __shared__ float s8[8];
  const unsigned T = scal_u[0];
  const int i = blockIdx.x * 256 + threadIdx.x;  // < BA (exact grid)
  const float v = ce_buf[i];
  float s = 0.f;
  if (v >= 0.f) {
    const unsigned bits = __float_as_uint(v);
    const unsigned bin = bits >> 21;
    if (bin > T) s = v;
    else if (bin == T) atomicAdd(&g_sub[(bits >> 10) & 0x7FFu], 1u);
  }
  s = wave_sum(s);
  if ((threadIdx.x & 31) == 0) s8[threadIdx.x >> 5] = s;
  __syncthreads();
  if (threadIdx.x == 0) {
    float r = 0.f;
#pragma unroll
    for (int w = 0; w < 8; ++w) r += s8[w];
    partB[blockIdx.x] = r;
  }
}

// ---------------- K4: reduce sum-above, pick fine threshold sub-bin S --------
__global__ __launch_bounds__(256) void k4_select2(
    const float* __restrict__ partB, const unsigned* __restrict__ g_sub,
    unsigned* __restrict__ scal_u, float* __restrict__ scal_f) {
  __shared__ float    r0[256];
  __shared__ unsigned s_cnt[HBINS];
  __shared__ int      s_suf[256];
  __shared__ unsigned s_S, s_R2;
  const int t = threadIdx.x;

  float sa = 0.f;
  for (int b = t; b < K3_BLOCKS; b += 256) sa += partB[b];
  r0[t] = sa;
  for (int i = t; i < HBINS; i += 256) s_cnt[i] = g_sub[i];
  __syncthreads();

  int csum = 0;
#pragma unroll
  for (int i = 0; i < 8; ++i) csum += (int)s_cnt[8 * t + i];
  s_suf[t] = csum;
  __syncthreads();

  if (t == 0) {
    float a = 0.f;
    for (int i = 0; i < 256; ++i) a += r0[i];
    scal_f[3] = a;  // sum above bin T
    int suf = 0;
    for (int i = 255; i >= 0; --i) { int cs = s_suf[i]; s_suf[i] = suf; suf += cs; }
    s_S = 0xFFFFu; s_R2 = 0u;
  }
  __syncthreads();

  const int R = (int)scal_u[1];
  if (R > 0) {
    const int above = s_suf[t];
    if (above < R && above + csum >= R) {
      int cum = above;
      for (int i = 7; i >= 0; --i) {
        const int c = (int)s_cnt[8 * t + i];
        if (cum + c >= R) { s_S = (unsigned)(8 * t + i); s_R2 = (unsigned)(R - cum); break; }
        cum += c;
      }
    }
  }
  __syncthreads();
  if (t == 0) { scal_u[2] = s_S; scal_u[3] = s_R2; }
}

// ---------------- K5: sum CE in bin T with sub-bin > S -----------------------
__global__ __launch_bounds__(256) void k5_mid(
    const float* __restrict__ ce_buf, const unsigned* __restrict__ scal_u,
    float* __restrict__ partC) {
  __shared__ float s8[8];
  const unsigned T = scal_u[0], S = scal_u[2];
  const int i = blockIdx.x * 256 + threadIdx.x;
  const float v = ce_buf[i];
  float s = 0.f;
  if (v >= 0.f) {
    const unsigned bits = __float_as_uint(v);
    if ((bits >> 21) == T && ((bits >> 10) & 0x7FFu) > S) s = v;
  }
  s = wave_sum(s);
  if ((threadIdx.x & 31) == 0) s8[threadIdx.x >> 5] = s;
  __syncthreads();
  if (threadIdx.x == 0) {
    float r = 0.f;
#pragma unroll
    for (int w = 0; w < 8; ++w) r += s8[w];
    partC[blockIdx.x] = r;
  }
}

// ---------------- K6: finalize ----------------------------------------------
__global__ __launch_bounds__(256) void k6_final(
    const float* __restrict__ partC, const unsigned* __restrict__ scal_u,
    const float* __restrict__ scal_f, float* __restrict__ out) {
  __shared__ float r0[256];
  const int t = threadIdx.x;
  float sm = 0.f;
  for (int b = t; b < K3_BLOCKS; b += 256) sm += partC[b];
  r0[t] = sm;
  __syncthreads();
  if (t == 0) {
    float acc = 0.f;
    for (int i = 0; i < 256; ++i) acc += r0[i];
    const unsigned T = scal_u[0], S = scal_u[2], R2 = scal_u[3];
    float residual = 0.f;
    if (R2 && T < HBINS && S < HBINS)  // boundary ties known to 22 bits
      residual = (float)R2 * __uint_as_float((T << 21) | (S << 10));
    const float np = scal_f[0], ll = scal_f[1], pc = scal_f[2], sa = scal_f[3];
    out[0] = (ll + pc + sa + acc + residual) / np;
  }
}

// ---------------- host-side launcher ----------------------------------------
extern "C" void kernel_launch(void* const* d_in, const int* in_sizes, int n_in,
                              void* d_out, int out_size, void* d_ws,
                              size_t ws_size, hipStream_t stream) {
  const float* pred_loc  = (const float*)d_in[0];
  const float* pred_conf = (const float*)d_in[1];
  const float* gt_loc    = (const float*)d_in[2];
  const int*   gt_label  = (const int*)d_in[3];
  float* out = (float*)d_out;

  float*    ws     = (float*)d_ws;
  float*    ce_buf = ws;                                 // BA floats
  unsigned* g_hist = (unsigned*)(ws + BA);               // 2048 u32
  unsigned* g_sub  = g_hist + HBINS;                     // 2048 u32
  float*    partA  = (float*)(g_sub + HBINS);            // 3*K1_BLOCKS
  float*    partB  = partA + 3 * K1_BLOCKS;              // K3_BLOCKS
  float*    partC  = partB + K3_BLOCKS;                  // K3_BLOCKS
  unsigned* scal   = (unsigned*)(partC + K3_BLOCKS);     // 16 u32
  float*    scalf  = (float*)(scal + 8);

  size_t need = (size_t)((char*)(scal + 16) - (char*)d_ws);
  if (ws_size < need) return;  // ~2.3 MB required

  k0_init<<<8, 256, 0, stream>>>(g_hist, g_sub, scal);
  k1_main<<<K1_BLOCKS, 256, 0, stream>>>(pred_loc, pred_conf, gt_loc, gt_label,
                                         ce_buf, g_hist, partA);
  k2_select<<<1, 256, 0, stream>>>(partA, g_hist, scal, scalf);
  k3_above<<<K3_BLOCKS, 256, 0, stream>>>(ce_buf, scal, g_sub, partB);
  k4_select2<<<1, 256, 0, stream>>>(partB, g_sub, scal, scalf);
  k5_mid<<<K3_BLOCKS, 256, 0, stream>>>(ce_buf, scal, partC);
  k6_final<<<1, 256, 0, stream>>>(partC, scal, scalf, out);
}